// Gate_81647328297466
// MI455X (gfx1250) — compile-verified
//
#include <hip/hip_runtime.h>

typedef __attribute__((ext_vector_type(2))) float v2f;
typedef __attribute__((ext_vector_type(4))) float v4f;
typedef __attribute__((ext_vector_type(8))) float v8f;

#define NUM_TOKENS   16384
#define D_MODEL      2048
#define NUM_EXPERTS  8
#define KC           64                 // k-chunk staged through LDS
#define ROW_STRIDE   (KC + 4)           // 68 floats -> conflict-free wmma A reads
#define WAVES_PER_BLOCK 4
#define TOKENS_PER_WAVE 16
#define TOKENS_PER_BLOCK (WAVES_PER_BLOCK * TOKENS_PER_WAVE)

// Merge two descending-sorted top-2 lists (value desc, index asc on ties —
// matches jax.lax.top_k tie-breaking).
__device__ __forceinline__ void top2_merge(float &v1, int &i1, float &v2, int &i2,
                                           float w1, int j1, float w2, int j2) {
  bool firstOurs = (v1 > w1) || ((v1 == w1) && (i1 < j1));
  float a   = firstOurs ? v2 : v1;  int ia = firstOurs ? i2 : i1;
  float b   = firstOurs ? w1 : w2;  int ib = firstOurs ? j1 : j2;
  float nv1 = firstOurs ? v1 : w1;  int ni1 = firstOurs ? i1 : j1;
  bool secOurs = (a > b) || ((a == b) && (ia < ib));
  v2 = secOurs ? a : b;  i2 = secOurs ? ia : ib;
  v1 = nv1;              i1 = ni1;
}

__global__ __launch_bounds__(WAVES_PER_BLOCK * 32)
void moe_gate_topk(const float* __restrict__ x, const float* __restrict__ W,
                   float* __restrict__ rw_out, float* __restrict__ idx_out) {
  // W^T pair-interleaved: lds_w[(k>>1)*16 + n*2 + (k&1)] = W[n][k]  (64 KB)
  __shared__ float lds_w[(D_MODEL / 2) * 16];
  // per-wave 16 x KC fp32 tile, padded row stride (17 KB)
  __shared__ float lds_a[WAVES_PER_BLOCK][TOKENS_PER_WAVE * ROW_STRIDE];

  const int tid  = threadIdx.x;
  const int wave = tid >> 5;
  const int lane = tid & 31;
  const int l15  = lane & 15;
  const int half = lane >> 4;      // selects K sub-pair for A/B fragments

  // ---- one-time W stage (coalesced global reads) ----
  for (int i = tid; i < NUM_EXPERTS * D_MODEL; i += WAVES_PER_BLOCK * 32) {
    int n = i >> 11;               // i / D_MODEL
    int k = i & (D_MODEL - 1);
    lds_w[(k >> 1) * 16 + n * 2 + (k & 1)] = W[i];
  }
  __syncthreads();

  const int row0 = blockIdx.x * TOKENS_PER_BLOCK + wave * TOKENS_PER_WAVE;
  float* aw = (float*)lds_a[wave];

  v8f c = {0.f, 0.f, 0.f, 0.f, 0.f, 0.f, 0.f, 0.f};

  for (int kc = 0; kc < D_MODEL; kc += KC) {
    // wave-local coalesced staging: 16 rows x 64 cols, 256B contiguous per row,
    // non-temporal (x is single-use; don't thrash caches)
#pragma unroll
    for (int it = 0; it < 8; ++it) {
      int linear = it * 128 + lane * 4;
      int r   = linear >> 6;
      int col = linear & 63;
      const v4f* src = (const v4f*)(x + (size_t)(row0 + r) * D_MODEL + kc + col);
      v4f v = __builtin_nontemporal_load(src);
      *(v4f*)(aw + r * ROW_STRIDE + col) = v;   // 2-way store conflict = minimum for 512B
    }
    if (kc + KC < D_MODEL) {   // global_prefetch_b8 for the next chunk's lines
      __builtin_prefetch(x + (size_t)(row0 + l15) * D_MODEL + kc + KC + half * 32, 0, 1);
    }
    __builtin_amdgcn_wave_barrier();  // cross-lane LDS RAW; HW DS pipe is in-order per wave

#pragma unroll
    for (int kk = 0; kk < KC; kk += 4) {
      // A fragment (16x4 f32): lane m = l15, VGPR pair holds K = kk+2*half, +1
      v2f a = *(const v2f*)(aw + l15 * ROW_STRIDE + kk + half * 2);
      // B fragment (4x16 f32): lane n, K pair selected by half; lanes 8..15
      // replay expert columns 0..7 (broadcast read, outputs N>=8 unused)
      int q = ((kc + kk) >> 1) + half;
      v2f b = *(const v2f*)(lds_w + q * 16 + (l15 & 7) * 2);
      c = __builtin_amdgcn_wmma_f32_16x16x4_f32(false, a, false, b,
                                                (short)0, c, false, false);
    }
    __builtin_amdgcn_wave_barrier();  // WAR vs next chunk's staging stores
  }

  // ---- epilogue: C VGPR j holds token m = j + 8*half, expert n = l15 ----
#pragma unroll
  for (int j = 0; j < 8; ++j) {
    float v1 = c[j];                 int i1 = l15;
    float v2 = -__builtin_inff();    int i2 = NUM_EXPERTS;
#pragma unroll
    for (int mask = 1; mask <= 4; mask <<= 1) {   // butterfly over 8-lane expert group
      float ov1 = __shfl_xor(v1, mask, 32);
      int   oi1 = __shfl_xor(i1, mask, 32);
      float ov2 = __shfl_xor(v2, mask, 32);
      int   oi2 = __shfl_xor(i2, mask, 32);
      top2_merge(v1, i1, v2, i2, ov1, oi1, ov2, oi2);
    }
    if (l15 == 0) {                  // lanes 0 and 16 publish their half's token
      float t   = __expf(v2 - v1);   // <= 1, softmax denom cancels after renorm
      float inv = 1.0f / (1.0f + t);
      int tok = row0 + half * 8 + j;
      rw_out[2 * tok]      = inv;
      rw_out[2 * tok + 1]  = t * inv;
      idx_out[2 * tok]     = (float)i1;
      idx_out[2 * tok + 1] = (float)i2;
    }
  }
}

extern "C" void kernel_launch(void* const* d_in, const int* in_sizes, int n_in,
                              void* d_out, int out_size, void* d_ws, size_t ws_size,
                              hipStream_t stream) {
  (void)in_sizes; (void)n_in; (void)out_size; (void)d_ws; (void)ws_size;
  const float* x = (const float*)d_in[0];
  const float* W = (const float*)d_in[1];
  float* rw  = (float*)d_out;                 // [16384, 2] routing weights
  float* idx = rw + 2 * NUM_TOKENS;           // [16384, 2] expert ids (as float)
  dim3 grid(NUM_TOKENS / TOKENS_PER_BLOCK);   // 256 blocks
  dim3 block(WAVES_PER_BLOCK * 32);           // 128 threads = 4 waves
  moe_gate_topk<<<grid, block, 0, stream>>>(x, W, rw, idx);
}